// Choice_73143293051370
// MI455X (gfx1250) — compile-verified
//
#include <hip/hip_runtime.h>
#include <cstdint>

#define N_EXPERTS 8
#define CHANNELS  512
#define BATCH     8
#define HW        3136            // 56*56
#define TILE      512             // floats per expert row per TDM tile
#define NBUF      4               // LDS ring: 4 x 16KB = 64KB
#define DEPTH     3               // TDM loads kept in flight
#define THREADS   128             // 4 waves; one float4 per lane per tile
#define NTILES    7               // 6*512 + 64 = 3136
#define BUFBYTES  (N_EXPERTS * TILE * 4)

typedef float    f4  __attribute__((ext_vector_type(4)));
typedef uint32_t v4u __attribute__((ext_vector_type(4)));
typedef uint32_t v8u __attribute__((ext_vector_type(8)));

// Expert stride inside l_learner[E,B,C,H,W], in elements.
#define E_STRIDE 12845056ull      // BATCH*CHANNELS*HW

// Issue one TDM load of an (8 x tlen)-float 2D tile into LDS.
// D# per CDNA5 ISA ch.8: group0 (count/lds_addr/global_addr/type=2),
// group1 (data_size=4B, tensor_dim0/1, tile_dim0/1, dim0-stride);
// groups 2/3 omitted -> 2D tensor mode. Tracked by TENSORcnt.
__device__ __forceinline__ void tdm_load_tile(uint32_t lds_off, uint64_t gaddr,
                                              uint32_t tlen) {
  v4u g0;
  g0[0] = 1u;                                       // count=1, user mode, no gather
  g0[1] = lds_off;                                  // lds_addr [63:32]
  g0[2] = (uint32_t)gaddr;                          // global_addr [95:64]
  g0[3] = ((uint32_t)(gaddr >> 32) & 0x01FFFFFFu)   // global_addr [120:96]
          | (2u << 30);                             // type=2 ("image") [127:126]

  v8u g1;
  g1[0] = (2u << 16);                               // wg_mask=0, data_size=2 (4B)
  g1[1] = (tlen & 0xFFFFu) << 16;                   // abar_addr=0 | tensor_dim0[15:0]
  g1[2] = ((tlen >> 16) & 0xFFFFu)                  // tensor_dim0[31:16]
          | (8u << 16);                             // tensor_dim1[15:0] = 8 experts
  g1[3] = (tlen & 0xFFFFu) << 16;                   // tensor_dim1[31:16]=0 | tile_dim0
  g1[4] = 8u;                                       // tile_dim1 = 8, tile_dim2 = 0
  g1[5] = (uint32_t)E_STRIDE;                       // tensor_dim0_stride[31:0]
  g1[6] = (uint32_t)(E_STRIDE >> 32) & 0xFFFFu;     // stride[47:32] | dim1_stride=0
  g1[7] = 0u;                                       // dim1_stride hi = 0 (2D tile)

  asm volatile("tensor_load_to_lds %0, %1" :: "s"(g0), "s"(g1) : "memory");
}

__global__ void __launch_bounds__(THREADS)
moe_gate_wsum_kernel(const float* __restrict__ L,   // [E,B,C,H,W]
                     const float* __restrict__ G,   // [B, E*C]
                     float* __restrict__ out) {     // [B,C,H,W]
  __shared__ float lds[NBUF][N_EXPERTS * TILE];     // 64 KB ring

  const int plane = blockIdx.x;                     // b*CHANNELS + c
  const int b   = plane >> 9;
  const int c   = plane & 511;
  const int tid = threadIdx.x;
  const bool leader = (tid < 32);                   // wave 0 drives the TDM

  // Per-(b,c) gates: g[b, e*512 + c] — uniform, lands in SGPRs via s_load.
  float gate[N_EXPERTS];
#pragma unroll
  for (int e = 0; e < N_EXPERTS; ++e)
    gate[e] = G[(size_t)b * (N_EXPERTS * CHANNELS) + e * CHANNELS + c];

  const uint64_t lbytes =
      (uint64_t)(uintptr_t)L + ((uint64_t)plane * HW) * 4ull;
  const uint32_t lds_base = (uint32_t)(uintptr_t)(&lds[0][0]);

  // ---- Prologue: fill the pipeline with DEPTH tile loads (all full TILEs,
  // since NTILES > DEPTH), then make tile 0 visible to all waves.
  if (leader) {
#pragma unroll
    for (int p = 0; p < DEPTH; ++p)
      tdm_load_tile(lds_base + p * BUFBYTES,
                    lbytes + (uint64_t)(p * TILE) * 4ull, (uint32_t)TILE);
    __builtin_amdgcn_s_wait_tensorcnt(DEPTH - 1);   // tile 0 landed
  }
  __syncthreads();

  for (int t = 0; t < NTILES; ++t) {
    const int tstart = t * TILE;
    const int tlen   = (HW - tstart) < TILE ? (HW - tstart) : TILE;

    // Keep DEPTH loads in flight: stage tile t+DEPTH into buf (t+DEPTH)&3.
    // That buffer's last consumer (tile t-1) finished at the previous barrier.
    if (leader && (t + DEPTH) < NTILES) {
      const int nstart = (t + DEPTH) * TILE;
      const int nlen   = (HW - nstart) < TILE ? (HW - nstart) : TILE;
      tdm_load_tile(lds_base + ((t + DEPTH) & (NBUF - 1)) * BUFBYTES,
                    lbytes + (uint64_t)nstart * 4ull, (uint32_t)nlen);
    }

    // Gate-weighted sum over 8 experts; one float4 per lane via ds_load_b128.
    const float* bufp = &lds[t & (NBUF - 1)][0];
    const int nvec = tlen >> 2;
    if (tid < nvec) {
      f4 acc = {0.f, 0.f, 0.f, 0.f};
#pragma unroll
      for (int e = 0; e < N_EXPERTS; ++e) {
        const f4 v = *reinterpret_cast<const f4*>(bufp + e * tlen + (tid << 2));
        acc += v * gate[e];
      }
      // Output is never re-read on-chip: bypass caches with NT b128 store.
      __builtin_nontemporal_store(
          acc,
          reinterpret_cast<f4*>(out + (size_t)plane * HW + tstart) + tid);
    }

    // Retire enough TDM ops (in-order) so tile t+1 is resident before the
    // next iteration. s_wait_tensorcnt takes an immediate -> literal ladder.
    if (leader) {
      if (t < NTILES - DEPTH)            // loads t+1..t+DEPTH outstanding
        __builtin_amdgcn_s_wait_tensorcnt(DEPTH - 1);
      else if (t == NTILES - DEPTH)      // pipeline draining
        __builtin_amdgcn_s_wait_tensorcnt(1);
      else if (t == NTILES - 2)
        __builtin_amdgcn_s_wait_tensorcnt(0);
      // t == NTILES-1: nothing outstanding
    }
    __syncthreads();
  }
}

extern "C" void kernel_launch(void* const* d_in, const int* in_sizes, int n_in,
                              void* d_out, int out_size, void* d_ws,
                              size_t ws_size, hipStream_t stream) {
  const float* L = (const float*)d_in[0];   // l_learner [8,8,512,56,56] f32
  const float* G = (const float*)d_in[1];   // g         [8,4096]       f32
  float* out     = (float*)d_out;           // [8,512,56,56] f32
  (void)in_sizes; (void)n_in; (void)out_size; (void)d_ws; (void)ws_size;

  dim3 grid(BATCH * CHANNELS);              // 4096 planes, one per workgroup
  dim3 block(THREADS);
  hipLaunchKernelGGL(moe_gate_wsum_kernel, grid, block, 0, stream, L, G, out);
}